// Pinv_8272107012847
// MI455X (gfx1250) — compile-verified
//
#include <hip/hip_runtime.h>
#include <hip/hip_bf16.h>

typedef __attribute__((ext_vector_type(2))) float v2f;
typedef __attribute__((ext_vector_type(4))) float v4f;
typedef __attribute__((ext_vector_type(8))) float v8f;
typedef __attribute__((ext_vector_type(4))) unsigned int u32x4;
typedef __attribute__((ext_vector_type(4))) int i32x4;
typedef __attribute__((ext_vector_type(8))) int i32x8;

#define LDA 68  // padded LDS row stride (floats): 64 data + 4 pad; 272B = 17*16B

#if __has_builtin(__builtin_amdgcn_tensor_load_to_lds)
#define USE_TDM 1
#else
#define USE_TDM 0
#endif

__device__ __forceinline__ v8f wmma4(v2f a, v2f b, v8f c) {
  // 8 args: (neg_a, A, neg_b, B, c_mod, C, reuse_a, reuse_b) -> v_wmma_f32_16x16x4_f32
  return __builtin_amdgcn_wmma_f32_16x16x4_f32(false, a, false, b, (short)0, c, false, false);
}

// One workgroup (8 wave32) per batch matrix.
// A: 128x64 row-major.  out: 64x128 row-major = pinv(A) = (A^T A)^-1 A^T.
__global__ __launch_bounds__(256) void pinv_kernel(const float* __restrict__ Ag,
                                                   float* __restrict__ outg,
                                                   int batch) {
  __shared__ float As[128][LDA];     // A (keep FIRST in LDS so TDM descriptor sees low offset)
  __shared__ float Gs[64][LDA];      // G = A^T A
  __shared__ float Yb[2][64][LDA];   // Newton iterate (ping-pong)
  __shared__ float Ts[64][LDA];      // T = G*Y
  __shared__ float nrm_s;

  const int b    = blockIdx.x;
  const int tid  = threadIdx.x;
  const int lane = tid & 31;
  const int wave = tid >> 5;
  const int nlo  = lane & 15;   // M or N index within 16-wide tile
  const int hi   = lane >> 4;   // half-wave select
  const int kb   = hi * 2;      // K offset within the 4-chunk (VGPR0 K=kb, VGPR1 K=kb+1)

  if (b >= batch) return;  // workgroup-uniform

  const float* Ab = Ag + (size_t)b * (128 * 64);

  // ---------------- stage A (128x64) into LDS with padded stride ----------------
#if USE_TDM
  if (tid < 32) {  // one wave issues the TDM DMA (EXEC ignored by tensor ops)
    const uint64_t ga       = (uint64_t)(uintptr_t)Ab;
    const uint32_t lds_base = (uint32_t)(uintptr_t)(void*)&As[0][0];  // flat low 32b = LDS addr
    u32x4 g0;
    g0[0] = 1u;                                      // count=1 (valid user descriptor)
    g0[1] = lds_base;                                // lds_addr
    g0[2] = (uint32_t)(ga & 0xFFFFFFFFu);            // global_addr[31:0]
    g0[3] = (uint32_t)((ga >> 32) & 0x01FFFFFFu)     // global_addr[56:32]
          | (2u << 30);                              // type = 2 ("image")
    i32x8 g1;
    g1[0] = (2 << 16)      // data_size = 4B
          | (1 << 20)      // pad_enable (LDS dest padding)
          | (5 << 22)      // pad_interval: every 64 DWORDs (one 64-float row)
          | (3 << 25);     // pad_amount: 4 DWORDs  -> LDS row stride 68 floats
    g1[1] = (64  << 16);   // tensor_dim0 = 64  (row length, elements)
    g1[2] = (128 << 16);   // tensor_dim1 = 128 (rows)
    g1[3] = (64  << 16);   // tile_dim0 = 64
    g1[4] = 128;           // tile_dim1 = 128
    g1[5] = 64;            // tensor_dim0_stride = 64
    g1[6] = 0;
    g1[7] = 0;
    i32x4 gz4 = {0, 0, 0, 0};                        // 2D tensor: groups 2/3 unused
    i32x8 gz8 = {0, 0, 0, 0, 0, 0, 0, 0};
    __builtin_amdgcn_tensor_load_to_lds(g0, g1, gz4, gz4, gz8, 0);
    __builtin_amdgcn_s_wait_tensorcnt(0);            // issuing wave waits for DMA completion
  }
  __syncthreads();
#else
  for (int i = 0; i < 8; ++i) {
    int idx = tid + 256 * i;           // 2048 float4's total
    int e   = idx * 4;
    int row = e >> 6;
    int col = e & 63;
    v4f val = *(const v4f*)(Ab + e);
    *(v4f*)(&As[row][col]) = val;
  }
  __syncthreads();
#endif

  // ---------------- G = A^T A : 16 tiles of 16x16, K = 128 ----------------
  for (int t = wave * 2; t < wave * 2 + 2; ++t) {
    int ti = t >> 2, tj = t & 3;
    v8f acc = {};
    for (int kk = 0; kk < 128; kk += 4) {
      v2f a, bb;
      a.x  = As[kk + kb][ti * 16 + nlo];        // (A^T)[m][k] = A[k][ti*16+m]
      a.y  = As[kk + kb + 1][ti * 16 + nlo];
      bb.x = As[kk + kb][tj * 16 + nlo];        // A[k][tj*16+n]
      bb.y = As[kk + kb + 1][tj * 16 + nlo];
      acc = wmma4(a, bb, acc);
    }
    int r0 = ti * 16 + hi * 8;
    for (int v = 0; v < 8; ++v) Gs[r0 + v][tj * 16 + nlo] = acc[v];
  }
  __syncthreads();

  // ---------------- alpha = 1/||G||_F  (<= 1/sigma_max, tighter than 1/trace) ----------------
  if (tid == 0) nrm_s = 0.0f;
  __syncthreads();
  float part = 0.0f;
  for (int i = tid; i < 64 * 64; i += 256) {
    float g = Gs[i >> 6][i & 63];
    part += g * g;
  }
  atomicAdd(&nrm_s, part);
  __syncthreads();
  const float alpha = rsqrtf(nrm_s);

  // Y0 = alpha * I
  for (int i = tid; i < 64 * 64; i += 256) {
    int r = i >> 6, c = i & 63;
    Yb[0][r][c] = (r == c) ? alpha : 0.0f;
  }
  __syncthreads();

  // ---------------- Newton-Schulz: Y <- 2Y - Y(GY) ----------------
  int cur = 0;
  const int NEWTON = 20;
  for (int it = 0; it < NEWTON; ++it) {
    // T = G * Ycur
    for (int t = wave * 2; t < wave * 2 + 2; ++t) {
      int ti = t >> 2, tj = t & 3;
      v8f acc = {};
      for (int kk = 0; kk < 64; kk += 4) {
        v2f a, bb;
        a.x  = Gs[ti * 16 + nlo][kk + kb];
        a.y  = Gs[ti * 16 + nlo][kk + kb + 1];
        bb.x = Yb[cur][kk + kb][tj * 16 + nlo];
        bb.y = Yb[cur][kk + kb + 1][tj * 16 + nlo];
        acc = wmma4(a, bb, acc);
      }
      int r0 = ti * 16 + hi * 8;
      for (int v = 0; v < 8; ++v) Ts[r0 + v][tj * 16 + nlo] = acc[v];
    }
    __syncthreads();
    // Ynext = 2*Ycur - Ycur * T
    for (int t = wave * 2; t < wave * 2 + 2; ++t) {
      int ti = t >> 2, tj = t & 3;
      v8f acc = {};
      for (int kk = 0; kk < 64; kk += 4) {
        v2f a, bb;
        a.x  = Yb[cur][ti * 16 + nlo][kk + kb];
        a.y  = Yb[cur][ti * 16 + nlo][kk + kb + 1];
        bb.x = Ts[kk + kb][tj * 16 + nlo];
        bb.y = Ts[kk + kb + 1][tj * 16 + nlo];
        acc = wmma4(a, bb, acc);
      }
      int r0 = ti * 16 + hi * 8;
      for (int v = 0; v < 8; ++v) {
        int rr = r0 + v, cc = tj * 16 + nlo;
        Yb[cur ^ 1][rr][cc] = 2.0f * Yb[cur][rr][cc] - acc[v];
      }
    }
    __syncthreads();
    cur ^= 1;
  }

  // ---------------- P = Ycur * A^T  -> out[b] (64 x 128) ----------------
  float* Ob = outg + (size_t)b * (64 * 128);
  for (int t = wave * 4; t < wave * 4 + 4; ++t) {
    int ti = t >> 3, tm = t & 7;   // ti: 0..3 rows of P, tm: 0..7 col-tiles of P
    v8f acc = {};
    for (int kk = 0; kk < 64; kk += 4) {
      v2f a, bb;
      a.x  = Yb[cur][ti * 16 + nlo][kk + kb];
      a.y  = Yb[cur][ti * 16 + nlo][kk + kb + 1];
      bb.x = As[tm * 16 + nlo][kk + kb];        // (A^T)[k][n] = A[tm*16+n][k]
      bb.y = As[tm * 16 + nlo][kk + kb + 1];
      acc = wmma4(a, bb, acc);
    }
    int r0 = ti * 16 + hi * 8;
    for (int v = 0; v < 8; ++v) {
      Ob[(size_t)(r0 + v) * 128 + tm * 16 + nlo] = acc[v];
    }
  }
}

extern "C" void kernel_launch(void* const* d_in, const int* in_sizes, int n_in,
                              void* d_out, int out_size, void* d_ws, size_t ws_size,
                              hipStream_t stream) {
  (void)n_in; (void)out_size; (void)d_ws; (void)ws_size;
  const float* A = (const float*)d_in[0];
  float* out = (float*)d_out;
  int batch = in_sizes[0] / (128 * 64);   // 4096
  pinv_kernel<<<batch, 256, 0, stream>>>(A, out, batch);
}